// QLSTM_65481071410098
// MI455X (gfx1250) — compile-verified
//
#include <hip/hip_runtime.h>
#include <hip/hip_bf16.h>

typedef float v2f __attribute__((ext_vector_type(2)));
typedef float v8f __attribute__((ext_vector_type(8)));

#define T_STEPS 256
#define BATCH   1024
#define IN_DIM  128
#define NQ      8
#define CAT     136   // IN_DIM + NQ
#define NA      32    // 4 gates * 8 qubits

// ---------------- setup: pack W[32][136], bias[32], cos/sin of RY params ----------------
__global__ void qlstm_setup(const float* __restrict__ Wf, const float* __restrict__ bf, const float* __restrict__ Pf,
                            const float* __restrict__ Wi, const float* __restrict__ bi, const float* __restrict__ Pi,
                            const float* __restrict__ Wg, const float* __restrict__ bg, const float* __restrict__ Pg,
                            const float* __restrict__ Wo, const float* __restrict__ bo, const float* __restrict__ Po,
                            float* __restrict__ Wall, float* __restrict__ bias, float* __restrict__ cps) {
    int tid = threadIdx.x;
    for (int idx = tid; idx < NA * CAT; idx += 256) {
        int n = idx / CAT, k = idx % CAT;
        const float* W = (n < 8) ? Wf : (n < 16) ? Wi : (n < 24) ? Wg : Wo;
        Wall[idx] = W[(n & 7) * CAT + k];
    }
    if (tid < NA) {
        const float* b = (tid < 8) ? bf : (tid < 16) ? bi : (tid < 24) ? bg : bo;
        bias[tid] = b[tid & 7];
    }
    if (tid < 64) {  // gate(2b) x (depth*8)
        int gt = tid >> 4, dw = tid & 15;
        const float* P = (gt == 0) ? Pf : (gt == 1) ? Pi : (gt == 2) ? Pg : Po;
        float p = 0.5f * P[dw];
        cps[(gt * 16 + dw) * 2 + 0] = cosf(p);
        cps[(gt * 16 + dw) * 2 + 1] = sinf(p);
    }
}

// ---------------- per-step angle GEMM: ang[B][32] = [x_t | h] @ Wall^T + b  (WMMA f32 16x16x4) ----------------
__global__ __launch_bounds__(64) void qlstm_angles(const float* __restrict__ x,
                                                   const float* __restrict__ hbuf,
                                                   const float* __restrict__ Wall,
                                                   const float* __restrict__ bias,
                                                   float* __restrict__ ang, int t) {
    const int lane = threadIdx.x & 31;
    const int nt   = threadIdx.x >> 5;     // 2 waves: N-tile 0/1
    const int m0   = blockIdx.x * 16;      // 64 M-tiles
    const int n0   = nt * 16;
    const int l15  = lane & 15;
    const int ksub = (lane >> 4) * 2;      // A/B f32 layout: half-wave holds K pairs {0,1} / {2,3}

    const float* xrow = x + ((size_t)t * BATCH + (m0 + l15)) * IN_DIM;
    const float* hrow = hbuf + (size_t)(m0 + l15) * NQ;
    const float* wrow = Wall + (size_t)(n0 + l15) * CAT;

    v8f acc;
    float bv = bias[n0 + l15];             // fold bias into accumulator (N = lane&15 in C layout)
    #pragma unroll
    for (int r = 0; r < 8; r++) acc[r] = bv;

    #pragma unroll
    for (int kc = 0; kc < 32; kc++) {      // K = 0..127 from x_t
        int k = kc * 4 + ksub;
        v2f a = *(const v2f*)(xrow + k);
        v2f b = *(const v2f*)(wrow + k);
        acc = __builtin_amdgcn_wmma_f32_16x16x4_f32(false, a, false, b, (short)0, acc, false, false);
    }
    #pragma unroll
    for (int kc = 32; kc < 34; kc++) {     // K = 128..135 from h
        int k = kc * 4 + ksub;
        v2f a = *(const v2f*)(hrow + (k - IN_DIM));
        v2f b = *(const v2f*)(wrow + k);
        acc = __builtin_amdgcn_wmma_f32_16x16x4_f32(false, a, false, b, (short)0, acc, false, false);
    }
    #pragma unroll
    for (int r = 0; r < 8; r++) {          // C/D: M = r + 8*(lane>>4), N = lane&15
        int M = (lane >> 4) * 8 + r;
        ang[(size_t)(m0 + M) * NA + (n0 + l15)] = acc[r];
    }
}

// ---------------- per-step quantum sim + LSTM update ----------------
// 1 wave = 1 (sample, gate) statevector sim; 8 waves/block = 2 samples.
// Amp index k = lane*8 + r  ->  wires 0..4 = lane bits 4..0 (shfl_xor), wires 5..7 = reg bits 2..0.
__global__ __launch_bounds__(256) void qlstm_sim(const float* __restrict__ ang,
                                                 const float* __restrict__ cps,
                                                 float* __restrict__ hbuf, float* __restrict__ cbuf,
                                                 float* __restrict__ outs,
                                                 float* __restrict__ hx_out, float* __restrict__ cx_out,
                                                 int t, int last) {
    __shared__ float zsh[2][4][NQ];
    const int tid  = threadIdx.x;
    const int lane = tid & 31;
    const int wave = tid >> 5;
    const int sl   = wave >> 2;
    const int gate = wave & 3;
    const int s    = blockIdx.x * 2 + sl;

    // RX half-angle sin/cos (8 angles, redundant per lane — broadcast loads)
    float cA[8], sA[8];
    const float* arow = ang + (size_t)s * NA + gate * NQ;
    #pragma unroll
    for (int w = 0; w < 8; w++) __sincosf(0.5f * arow[w], &sA[w], &cA[w]);

    // RY params for this gate
    float cp[2][8], sp[2][8];
    const float* cg = cps + gate * 32;
    #pragma unroll
    for (int d = 0; d < 2; d++)
        #pragma unroll
        for (int w = 0; w < 8; w++) {
            cp[d][w] = cg[(d * 8 + w) * 2 + 0];
            sp[d][w] = cg[(d * 8 + w) * 2 + 1];
        }

    // Product state after RX layer: amp(k) = (prod cos/sin) * (-i)^popcount(k)
    float re[8], im[8];
    #pragma unroll
    for (int r = 0; r < 8; r++) {
        int k = lane * 8 + r;
        float m = 1.f;
        #pragma unroll
        for (int w = 0; w < 8; w++) m *= ((k >> (7 - w)) & 1) ? sA[w] : cA[w];
        int pop = __popc(k) & 3;
        re[r] = (pop == 0) ? m : ((pop == 2) ? -m : 0.f);
        im[r] = (pop == 1) ? -m : ((pop == 3) ? m : 0.f);
    }

    #pragma unroll
    for (int d = 0; d < 2; d++) {
        // RY on cross-lane wires 0..4 (lane bit 4-w)
        #pragma unroll
        for (int w = 0; w < 5; w++) {
            int lb = 4 - w;
            float c = cp[d][w];
            float sgn = ((lane >> lb) & 1) ? sp[d][w] : -sp[d][w];
            #pragma unroll
            for (int r = 0; r < 8; r++) {
                float pre = __shfl_xor(re[r], 1 << lb, 32);
                float pim = __shfl_xor(im[r], 1 << lb, 32);
                re[r] = c * re[r] + sgn * pre;
                im[r] = c * im[r] + sgn * pim;
            }
        }
        // RY on in-register wires 5..7 (reg bit 7-w)
        #pragma unroll
        for (int w = 5; w < 8; w++) {
            int rb = 7 - w;
            float c = cp[d][w], sv = sp[d][w];
            #pragma unroll
            for (int r0 = 0; r0 < 8; r0++) {
                if (r0 & (1 << rb)) continue;
                int r1 = r0 | (1 << rb);
                float a0r = re[r0], a1r = re[r1], a0i = im[r0], a1i = im[r1];
                re[r0] = c * a0r - sv * a1r;  re[r1] = sv * a0r + c * a1r;
                im[r0] = c * a0i - sv * a1i;  im[r1] = sv * a0i + c * a1i;
            }
        }
        // CNOT chain, wires (i, i+1): control bit 7-i, target bit 6-i
        #pragma unroll
        for (int i = 0; i < 4; i++) {        // both bits in lane index
            int cb = 4 - i, tb = 3 - i;
            bool ctl = (lane >> cb) & 1;
            #pragma unroll
            for (int r = 0; r < 8; r++) {
                float pre = __shfl_xor(re[r], 1 << tb, 32);
                float pim = __shfl_xor(im[r], 1 << tb, 32);
                re[r] = ctl ? pre : re[r];
                im[r] = ctl ? pim : im[r];
            }
        }
        {   // i=4: control lane bit0, target reg bit2
            bool ctl = lane & 1;
            #pragma unroll
            for (int r = 0; r < 4; r++) {
                float tr = re[r], ti = im[r];
                re[r]     = ctl ? re[r + 4] : re[r];
                im[r]     = ctl ? im[r + 4] : im[r];
                re[r + 4] = ctl ? tr : re[r + 4];
                im[r + 4] = ctl ? ti : im[r + 4];
            }
        }
        {   // i=5: control reg bit2 -> swap (4,6),(5,7)
            float tp;
            tp = re[4]; re[4] = re[6]; re[6] = tp;  tp = im[4]; im[4] = im[6]; im[6] = tp;
            tp = re[5]; re[5] = re[7]; re[7] = tp;  tp = im[5]; im[5] = im[7]; im[7] = tp;
        }
        {   // i=6: control reg bit1 -> swap (2,3),(6,7)
            float tp;
            tp = re[2]; re[2] = re[3]; re[3] = tp;  tp = im[2]; im[2] = im[3]; im[3] = tp;
            tp = re[6]; re[6] = re[7]; re[7] = tp;  tp = im[6]; im[6] = im[7]; im[7] = tp;
        }
    }

    // Z expectations
    float tot = 0.f, z5 = 0.f, z6 = 0.f, z7 = 0.f;
    #pragma unroll
    for (int r = 0; r < 8; r++) {
        float p = re[r] * re[r] + im[r] * im[r];
        tot += p;
        z5 += ((r >> 2) & 1) ? -p : p;
        z6 += ((r >> 1) & 1) ? -p : p;
        z7 += (r & 1) ? -p : p;
    }
    float z[8];
    #pragma unroll
    for (int w = 0; w < 5; w++) z[w] = ((lane >> (4 - w)) & 1) ? -tot : tot;
    z[5] = z5; z[6] = z6; z[7] = z7;
    #pragma unroll
    for (int off = 1; off < 32; off <<= 1)
        #pragma unroll
        for (int w = 0; w < 8; w++) z[w] += __shfl_xor(z[w], off, 32);

    if (lane == 0) {
        #pragma unroll
        for (int w = 0; w < 8; w++) zsh[sl][gate][w] = z[w];
    }
    __syncthreads();

    // LSTM pointwise update (16 threads: 2 samples x 8 qubits)
    if (tid < 16) {
        int sl2 = tid >> 3, q = tid & 7;
        int ss = blockIdx.x * 2 + sl2;
        float zf = zsh[sl2][0][q], zi = zsh[sl2][1][q], zg = zsh[sl2][2][q], zo = zsh[sl2][3][q];
        float f  = 1.f / (1.f + __expf(-zf));
        float ii = 1.f / (1.f + __expf(-zi));
        float g  = tanhf(zg);
        float o  = 1.f / (1.f + __expf(-zo));
        float c  = f * cbuf[ss * NQ + q] + ii * g;
        float h  = o * tanhf(c);
        cbuf[ss * NQ + q] = c;
        hbuf[ss * NQ + q] = h;
        outs[((size_t)t * BATCH + ss) * NQ + q] = h;
        if (last) { hx_out[ss * NQ + q] = h; cx_out[ss * NQ + q] = c; }
    }
}

extern "C" void kernel_launch(void* const* d_in, const int* in_sizes, int n_in,
                              void* d_out, int out_size, void* d_ws, size_t ws_size,
                              hipStream_t stream) {
    const float* x  = (const float*)d_in[0];
    const float* Wf = (const float*)d_in[1];  const float* bf = (const float*)d_in[2];  const float* Pf = (const float*)d_in[3];
    const float* Wi = (const float*)d_in[4];  const float* bi = (const float*)d_in[5];  const float* Pi = (const float*)d_in[6];
    const float* Wg = (const float*)d_in[7];  const float* bg = (const float*)d_in[8];  const float* Pg = (const float*)d_in[9];
    const float* Wo = (const float*)d_in[10]; const float* bo = (const float*)d_in[11]; const float* Po = (const float*)d_in[12];

    float* ws   = (float*)d_ws;
    float* ang  = ws;                        // 1024*32      = 32768 floats
    float* hbuf = ws + 32768;                // 1024*8       =  8192
    float* cbuf = ws + 40960;                // 1024*8       =  8192
    float* Wall = ws + 49152;                // 32*136       =  4352
    float* bias = ws + 53504;                // 32
    float* cps  = ws + 53536;                // 4*2*8*2      =   128

    float* outs = (float*)d_out;                       // [T,B,8]
    float* hx   = outs + (size_t)T_STEPS * BATCH * NQ; // [B,8]
    float* cx   = hx + (size_t)BATCH * NQ;             // [B,8]

    hipMemsetAsync(hbuf, 0, (size_t)BATCH * NQ * sizeof(float), stream);
    hipMemsetAsync(cbuf, 0, (size_t)BATCH * NQ * sizeof(float), stream);

    qlstm_setup<<<1, 256, 0, stream>>>(Wf, bf, Pf, Wi, bi, Pi, Wg, bg, Pg, Wo, bo, Po, Wall, bias, cps);

    for (int t = 0; t < T_STEPS; t++) {
        qlstm_angles<<<64, 64, 0, stream>>>(x, hbuf, Wall, bias, ang, t);
        qlstm_sim<<<512, 256, 0, stream>>>(ang, cps, hbuf, cbuf, outs, hx, cx, t, t == T_STEPS - 1);
    }
}